// Model_455266533973
// MI455X (gfx1250) — compile-verified
//
#include <hip/hip_runtime.h>

// CDNA5 (gfx1250) fused attention-with-residual-scores.
//   scores = q@k * scale + prev + attn_mask ; mask -> -inf
//   attn   = softmax(scores) ; out = attn @ v
// Outputs concatenated in d_out: [out | attn | scores].
//
// fp32 path via V_WMMA_F32_16X16X4_F32 (wave32). One workgroup of 128
// threads (4 waves) per (b, h, 16-row q tile). Score stripe (16x1024 f32,
// padded stride 1028 for LDS bank spread) lives in LDS across all phases.

typedef float v2f __attribute__((ext_vector_type(2)));
typedef float v8f __attribute__((ext_vector_type(8)));

#define SEQ   1024
#define HDIM  64
#define QT    16        // q rows per workgroup
#define SSTR  1028      // LDS stride of score stripe (1024 + 4 pad)
#define QSTR  68        // LDS stride of q tile (64 + 4 pad)

__global__ __launch_bounds__(128)
void attn_fused_f32_wmma(const float* __restrict__ q,      // [B,H,S,DK]
                         const float* __restrict__ k,      // [B,H,DK,S]
                         const float* __restrict__ v,      // [B,H,S,DK]
                         const float* __restrict__ prev,   // [B,H,S,S]
                         const float* __restrict__ amask,  // [1,S,S]
                         const unsigned char* __restrict__ kpm, // [B,S] bool
                         const float* __restrict__ scale_p,
                         float* __restrict__ out,          // [B,H,S,DK]
                         float* __restrict__ attn_o,       // [B,H,S,S]
                         float* __restrict__ scores_o)     // [B,H,S,S]
{
    __shared__ float q_lds[QT * QSTR];
    __shared__ float s_lds[QT * SSTR];

    const int q0 = blockIdx.x * QT;
    const int h  = blockIdx.y;
    const int b  = blockIdx.z;
    const size_t bh = (size_t)b * gridDim.y + h;

    const float* q_bh = q    + bh * (size_t)SEQ * HDIM;
    const float* k_bh = k    + bh * (size_t)HDIM * SEQ;
    const float* v_bh = v    + bh * (size_t)SEQ * HDIM;
    const float* p_bh = prev + bh * (size_t)SEQ * SEQ;
    float* o_bh  = out      + bh * (size_t)SEQ * HDIM;
    float* a_bh  = attn_o   + bh * (size_t)SEQ * SEQ;
    float* sc_bh = scores_o + bh * (size_t)SEQ * SEQ;

    const int tid  = threadIdx.x;
    const int wave = tid >> 5;
    const int lane = tid & 31;
    const int half = lane >> 4;   // selects K pair (A/B frags) / M+8 (C frag)
    const int lm   = lane & 15;   // M for A frag, N for B and C frags
    const float scale = scale_p[0];

    // ---- stage q tile (16 x 64) into LDS ----
    for (int i = tid; i < QT * HDIM; i += 128) {
        const int r = i >> 6, c = i & 63;
        q_lds[r * QSTR + c] = q_bh[(size_t)(q0 + r) * HDIM + c];
    }
    __syncthreads();

    // ---- GEMM1: scores stripe 16 x 1024, 64 col-tiles split over 4 waves ----
    for (int ct = wave; ct < SEQ / 16; ct += 4) {
        const int kcol0 = ct * 16;
        v8f acc = {};
        #pragma unroll
        for (int d = 0; d < HDIM; d += 4) {
            v2f a, bm;
            // A (16x4): lane holds rows lm, K = d + 2*half .. +1
            a.x = q_lds[lm * QSTR + d + 2 * half];
            a.y = q_lds[lm * QSTR + d + 2 * half + 1];
            // B (4x16): row-striped; k is [DK,S] so rows are contiguous in S
            const float* kp = k_bh + (size_t)(d + 2 * half) * SEQ + kcol0 + lm;
            bm.x = kp[0];
            bm.y = kp[SEQ];
            acc = __builtin_amdgcn_wmma_f32_16x16x4_f32(
                false, a, false, bm, (short)0, acc, false, false);
        }
        // epilogue: scale + prev + mask, key-padding -inf; emit scores + LDS
        #pragma unroll
        for (int j = 0; j < 8; ++j) {
            const int M  = j + 8 * half;
            const int qq = q0 + M;
            const int kk = kcol0 + lm;
            float s = acc[j] * scale
                    + p_bh[(size_t)qq * SEQ + kk]
                    + amask[(size_t)qq * SEQ + kk];
            if (kpm[(size_t)b * SEQ + kk]) s = -__builtin_inff();
            sc_bh[(size_t)qq * SEQ + kk] = s;
            s_lds[M * SSTR + kk] = s;
        }
    }
    __syncthreads();

    // ---- softmax over LDS stripe: each wave owns 4 rows ----
    for (int r = wave * 4; r < wave * 4 + 4; ++r) {
        float* row = &s_lds[r * SSTR];
        float mx = -__builtin_inff();
        for (int c = lane; c < SEQ; c += 32) mx = fmaxf(mx, row[c]);
        #pragma unroll
        for (int off = 16; off > 0; off >>= 1)
            mx = fmaxf(mx, __shfl_xor(mx, off, 32));

        float sum = 0.f;
        for (int c = lane; c < SEQ; c += 32) {
            const float e = __expf(row[c] - mx);
            row[c] = e;
            sum += e;
        }
        #pragma unroll
        for (int off = 16; off > 0; off >>= 1)
            sum += __shfl_xor(sum, off, 32);
        const float inv = 1.0f / sum;

        float* ag = &a_bh[(size_t)(q0 + r) * SEQ];
        for (int c = lane; c < SEQ; c += 32) {
            const float a = row[c] * inv;
            row[c] = a;
            ag[c]  = a;   // coalesced attn output
        }
    }
    __syncthreads();

    // ---- GEMM2: out tile 16 x 64; each wave owns one 16x16 d-tile, K=1024 ----
    const int d0 = wave * 16;
    v8f acc2 = {};
    for (int kk = 0; kk < SEQ; kk += 4) {
        v2f a, bm;
        a.x = s_lds[lm * SSTR + kk + 2 * half];
        a.y = s_lds[lm * SSTR + kk + 2 * half + 1];
        const float* vp = v_bh + (size_t)(kk + 2 * half) * HDIM + d0 + lm;
        bm.x = vp[0];
        bm.y = vp[HDIM];
        acc2 = __builtin_amdgcn_wmma_f32_16x16x4_f32(
            false, a, false, bm, (short)0, acc2, false, false);
    }
    #pragma unroll
    for (int j = 0; j < 8; ++j) {
        const int M = j + 8 * half;
        o_bh[(size_t)(q0 + M) * HDIM + d0 + lm] = acc2[j];
    }
}

extern "C" void kernel_launch(void* const* d_in, const int* in_sizes, int n_in,
                              void* d_out, int out_size, void* d_ws, size_t ws_size,
                              hipStream_t stream) {
    (void)in_sizes; (void)n_in; (void)out_size; (void)d_ws; (void)ws_size;
    const float*         q     = (const float*)d_in[0];
    const float*         k     = (const float*)d_in[1];
    const float*         v     = (const float*)d_in[2];
    const float*         prev  = (const float*)d_in[3];
    const float*         amask = (const float*)d_in[4];
    const unsigned char* kpm   = (const unsigned char*)d_in[5];
    const float*         scale = (const float*)d_in[6];

    float* out    = (float*)d_out;                                  // B*H*S*DK
    float* attn   = out  + (size_t)4 * 16 * 1024 * 64;              // B*H*S*S
    float* scores = attn + (size_t)4 * 16 * 1024 * 1024;            // B*H*S*S

    dim3 grid(SEQ / QT, 16, 4);   // (q tiles, H, B)
    attn_fused_f32_wmma<<<grid, 128, 0, stream>>>(
        q, k, v, prev, amask, kpm, scale, out, attn, scores);
}